// Bag_of_words_58609123721528
// MI455X (gfx1250) — compile-verified
//
#include <hip/hip_runtime.h>
#include <hip/hip_bf16.h>
#include <stdint.h>

#define VOCAB     30522
#define SEQ       512
#define HIST_PAD  30528   // multiple of 8 floats -> exact float4 zeroing (7632 chunks)
#define BLOCK     512

typedef int v2i __attribute__((ext_vector_type(2)));
typedef __attribute__((address_space(1))) v2i* gv2i_t;   // global, 2-dword pointee
typedef __attribute__((address_space(3))) v2i* lv2i_t;   // LDS, 2-dword pointee

static __device__ __forceinline__ void async_store_lds_b64(float* gdst, float* lsrc) {
#if defined(__has_builtin) && __has_builtin(__builtin_amdgcn_global_store_async_from_lds_b64)
    __builtin_amdgcn_global_store_async_from_lds_b64((gv2i_t)gdst, (lv2i_t)lsrc, 0, 0);
#else
    // VGLOBAL async encoding: VADDR = 64-bit global address, VSRC = LDS byte address
    unsigned lds_off = (unsigned)(uintptr_t)lsrc;
    asm volatile("global_store_async_from_lds_b64 %0, %1, off"
                 :: "v"(gdst), "v"(lds_off)
                 : "memory");
#endif
}

static __device__ __forceinline__ void wait_asynccnt0() {
#if defined(__has_builtin) && __has_builtin(__builtin_amdgcn_s_wait_asynccnt)
    __builtin_amdgcn_s_wait_asynccnt(0);
#else
    asm volatile("s_wait_asynccnt 0" ::: "memory");
#endif
}

// One workgroup per sentence row. Private f32 vocab histogram in LDS
// (122 KB -- fits because CDNA5 WGP has 320 KB LDS), accumulated with
// ds_add_f32 atomics, then streamed LDS -> HBM with async b64 stores.
__global__ __launch_bounds__(BLOCK)
void bow_hist_kernel(const int* __restrict__ ids, float* __restrict__ out) {
    extern __shared__ float hist[];       // HIST_PAD floats (dynamic LDS)
    __shared__ int s_first;               // index of first PAD token in this row

    const int tid = threadIdx.x;
    const int row = blockIdx.x;

    if (tid == 0) s_first = SEQ;

    // Zero the histogram with 16-byte LDS stores: 7632 float4 chunks.
    float4* hist4 = reinterpret_cast<float4*>(hist);
    const float4 z4 = make_float4(0.f, 0.f, 0.f, 0.f);
    for (int c = tid; c < HIST_PAD / 4; c += BLOCK) {
        hist4[c] = z4;                    // ds_store_b128
    }
    __syncthreads();

    // One thread per token (SEQ == BLOCK).
    const int tok = ids[(size_t)row * SEQ + tid];
    if (tok == 0) atomicMin(&s_first, tid);   // ds_min on LDS
    __syncthreads();

    const int first = s_first;
    if (tid < first) atomicAdd(&hist[tok], 1.0f);   // ds_add_f32, LDS-private
    __syncthreads();

    // Stream the finished histogram row to global memory.
    // 30522 floats = 15261 x 8-byte chunks exactly; row base (row*122088 B)
    // and LDS base are always 8-byte aligned, so b64 async stores are legal.
    float* dst = out + (size_t)row * VOCAB;
    const int NCHUNK = VOCAB / 2;   // 15261
    for (int c = tid; c < NCHUNK; c += BLOCK) {
        async_store_lds_b64(dst + 2 * c, hist + 2 * c);
    }

    // Drain ASYNCcnt before the wave releases its LDS.
    wait_asynccnt0();
}

extern "C" void kernel_launch(void* const* d_in, const int* in_sizes, int n_in,
                              void* d_out, int out_size, void* d_ws, size_t ws_size,
                              hipStream_t stream) {
    (void)n_in; (void)d_ws; (void)ws_size; (void)out_size;
    const int* ids = (const int*)d_in[0];
    float* out = (float*)d_out;

    const int B = in_sizes[0] / SEQ;          // 2048 rows
    const size_t lds_bytes = (size_t)HIST_PAD * sizeof(float);  // ~122 KB / WG

    bow_hist_kernel<<<dim3(B), dim3(BLOCK), lds_bytes, stream>>>(ids, out);
}